// S4Layer_10411000725867
// MI455X (gfx1250) — compile-verified
//
#include <hip/hip_runtime.h>

#define PI_F 3.14159265358979f
#define NTHREADS 256
#define NWAVES 8

typedef __attribute__((ext_vector_type(2))) float v2f;
typedef __attribute__((ext_vector_type(8))) float v8f;
typedef __attribute__((ext_vector_type(4))) unsigned v4u;
typedef __attribute__((ext_vector_type(4))) int v4i;
typedef __attribute__((ext_vector_type(8))) int v8i;

#if __has_builtin(__builtin_amdgcn_tensor_load_to_lds) && __has_builtin(__builtin_amdgcn_s_wait_tensorcnt)
#define HAVE_TDM 1
#else
#define HAVE_TDM 0
#endif

#if HAVE_TDM
// Issue one TDM descriptor: contiguous row of `nelem` f32 from global -> LDS.
// D# layout per CDNA5 ISA (2-group form, tensors <= 2D).
__device__ __forceinline__ void tdm_load_row_f32(const float* gsrc,
                                                 unsigned lds_byte_addr,
                                                 int nelem) {
  unsigned long long ga = (unsigned long long)(uintptr_t)gsrc;
  v4u g0;
  g0.x = 1u;                                            // count=1, user D#
  g0.y = lds_byte_addr;                                 // lds_addr
  g0.z = (unsigned)(ga & 0xFFFFFFFFu);                  // global_addr[31:0]
  g0.w = (unsigned)((ga >> 32) & 0x1FFFFFFu)            // global_addr[56:32]
       | (2u << 30);                                    // type=2 ("image")
  v8i g1;
  g1[0] = (int)(2u << 16);                              // data_size=2 (4 bytes)
  g1[1] = (int)((unsigned)nelem << 16);                 // tensor_dim0[15:0]
  g1[2] = (int)((((unsigned)nelem >> 16) & 0xFFFFu)     // tensor_dim0[31:16]
       | (1u << 16));                                   // tensor_dim1 = 1
  g1[3] = (int)((unsigned)nelem << 16);                 // tile_dim0 = nelem
  g1[4] = 1;                                            // tile_dim1=1, tile_dim2=0
  g1[5] = nelem;                                        // tensor_dim0_stride lo
  g1[6] = 0;                                            // stride hi / dim1_stride
  g1[7] = 0;
  v4i z4 = {};
#if defined(__clang_major__) && __clang_major__ >= 23
  v8i z8 = {};
  __builtin_amdgcn_tensor_load_to_lds(g0, g1, z4, z4, z8, 0);
#else
  __builtin_amdgcn_tensor_load_to_lds(g0, g1, z4, z4, 0);
#endif
}
#endif

// One 64x64x64 complex matmul  O = A * F  using V_WMMA_F32_16X16X4_F32 tiles.
// A element (r,k) at A[r*rsA + k*ksA]; F row-major (symmetric DFT-64 matrix).
// F_used = Fr + i*sgn*Fi  (sgn=+1 forward DFT, -1 inverse/conjugate).
// Output stored column-major: O[row + 64*col].
__device__ __forceinline__ void cmatmul64(
    const float* __restrict__ Ar, const float* __restrict__ Ai,
    int rsA, int ksA,
    const float* __restrict__ Fr, const float* __restrict__ Fi,
    float sgn, float scale,
    float* __restrict__ Or_, float* __restrict__ Oi_,
    int lane, int wave)
{
  const int mrow   = lane & 15;
  const int klo    = (lane < 16) ? 0 : 2;          // K split across lane halves
  const int rowoff = (lane >> 4) << 3;             // C/D: +8 rows in upper lanes
  for (int t = wave; t < 16; t += NWAVES) {
    const int ti = t >> 2, tj = t & 3;
    const int r = ti * 16 + mrow;                  // A fragment row
    const int c = tj * 16 + mrow;                  // B fragment col
    v8f accRR = {}; v8f accII = {}; v8f accRI = {}; v8f accIR = {};
    for (int k4 = 0; k4 < 16; ++k4) {
      const int kb = k4 * 4 + klo;
      v2f ar, ai, fr, fi;
      ar.x = Ar[r * rsA + kb * ksA];       ar.y = Ar[r * rsA + (kb + 1) * ksA];
      ai.x = Ai[r * rsA + kb * ksA];       ai.y = Ai[r * rsA + (kb + 1) * ksA];
      fr.x = Fr[kb * 64 + c];              fr.y = Fr[(kb + 1) * 64 + c];
      fi.x = Fi[kb * 64 + c];              fi.y = Fi[(kb + 1) * 64 + c];
      accRR = __builtin_amdgcn_wmma_f32_16x16x4_f32(false, ar, false, fr, (short)0, accRR, false, false);
      accII = __builtin_amdgcn_wmma_f32_16x16x4_f32(false, ai, false, fi, (short)0, accII, false, false);
      accRI = __builtin_amdgcn_wmma_f32_16x16x4_f32(false, ar, false, fi, (short)0, accRI, false, false);
      accIR = __builtin_amdgcn_wmma_f32_16x16x4_f32(false, ai, false, fr, (short)0, accIR, false, false);
    }
#pragma unroll
    for (int r8 = 0; r8 < 8; ++r8) {
      const int row = ti * 16 + r8 + rowoff;
      const int col = tj * 16 + mrow;
      Or_[row + 64 * col] = (accRR[r8] - sgn * accII[r8]) * scale;
      Oi_[row + 64 * col] = (sgn * accRI[r8] + accIR[r8]) * scale;
    }
  }
}

// 4096-point complex FFT, four-step (64x64). in==out allowed; tmp distinct.
// sgn=+1: forward DFT. sgn=-1 with scale=1/4096: jnp.fft.ifft convention.
__device__ __forceinline__ void fft4096(
    float* inr, float* ini, float* tr, float* ti_,
    float* outr, float* outi,
    const float* Fr, const float* Fi,
    float sgn, float scale, int tid)
{
  const int lane = tid & 31;
  const int wave = tid >> 5;
  __syncthreads();
  // Stage A: T[n1+64k1] = sum_n2 X[n1+64n2] * F[n2][k1]   (rsA=1, ksA=64)
  cmatmul64(inr, ini, 1, 64, Fr, Fi, sgn, 1.0f, tr, ti_, lane, wave);
  __syncthreads();
  // Twiddle: T[n1+64k1] *= exp(sgn * -2pi i * n1*k1 / 4096)
  for (int i = tid; i < 4096; i += NTHREADS) {
    int n1 = i & 63, k1 = i >> 6;
    float ang = sgn * (-2.0f * PI_F / 4096.0f) * (float)(n1 * k1);
    float s, c; __sincosf(ang, &s, &c);
    float xr = tr[i], xi = ti_[i];
    tr[i]  = xr * c - xi * s;
    ti_[i] = xr * s + xi * c;
  }
  __syncthreads();
  // Stage C: Out[k1+64k2] = sum_n1 T[n1+64k1] * F[n1][k2]  (rsA=64, ksA=1)
  cmatmul64(tr, ti_, 64, 1, Fr, Fi, sgn, scale, outr, outi, lane, wave);
  __syncthreads();
}

__global__ void __launch_bounds__(NTHREADS)
s4_channel_kernel(const float* __restrict__ u,
                  const float* __restrict__ Lre, const float* __restrict__ Lim,
                  const float* __restrict__ Pre, const float* __restrict__ Pim,
                  const float* __restrict__ Bre, const float* __restrict__ Bim,
                  const float* __restrict__ Cm,  const float* __restrict__ Dm,
                  const float* __restrict__ logstep,
                  float* __restrict__ out)
{
  extern __shared__ float sm[];
  float* Fr = sm;            // 4096: cos(2pi nk/64)
  float* Fi = Fr + 4096;     // 4096: -sin(2pi nk/64)
  float* Ar = Fi + 4096;     float* Ai = Ar + 4096;   // work A
  float* Br = Ai + 4096;     float* Bi = Br + 4096;   // fft tmp
  float* Sr = Bi + 4096;     float* Si = Sr + 4096;   // atRoots -> Ke
  float* Tr = Si + 4096;     float* Ti = Tr + 4096;   // work T (keeps yO)
  float* su = Ti + 4096;     // 4096: u staged via TDM
  float* lamr = su + 4096;   float* lami = lamr + 64;
  float* w00r = lami + 64;   float* w00i = w00r + 64;
  float* w01r = w00i + 64;   float* w01i = w01r + 64;
  float* w10r = w01i + 64;   float* w10i = w10r + 64;
  float* w11r = w10i + 64;   float* w11i = w11r + 64;

  const int h   = blockIdx.x;
  const int tid = threadIdx.x;
  const float step  = __expf(logstep[h]);
  const float dcoef = Dm[h];
  const float* uh = u + (size_t)h * 4096;
  float* oh = out + (size_t)h * 4096;

  // ---- async stage u[h,:] -> LDS via Tensor Data Mover (wave 0 issues) ----
#if HAVE_TDM
  if (tid < 32) {
    tdm_load_row_f32(uh, (unsigned)(uintptr_t)su, 4096);
  }
#else
  for (int l = tid; l < 4096; l += NTHREADS) su[l] = uh[l];
#endif

  // DFT-64 tables: F[n][k] = exp(-2pi i nk/64)  (symmetric)
  for (int i = tid; i < 4096; i += NTHREADS) {
    int nk = ((i >> 6) * (i & 63)) & 63;
    float ang = -2.0f * PI_F * (float)nk / 64.0f;
    float s, c; __sincosf(ang, &s, &c);
    Fr[i] = c; Fi[i] = s;
  }
  // Per-channel Cauchy numerators
  if (tid < 64) {
    const int n = tid, base = h * 64 + n;
    lamr[n] = fminf(Lre[base], -1e-4f);
    lami[n] = Lim[base];
    float cr = Cm[2 * base + 0], ci = Cm[2 * base + 1];
    float br = Bre[base], bi = Bim[base];
    float pr = Pre[base], pi = Pim[base];
    w00r[n] = cr * br + ci * bi;  w00i[n] = cr * bi - ci * br;  // conj(C)*B
    w01r[n] = cr * pr + ci * pi;  w01i[n] = cr * pi - ci * pr;  // conj(C)*P
    w10r[n] = pr * br + pi * bi;  w10i[n] = pr * bi - pi * br;  // conj(P)*B
    w11r[n] = pr * pr + pi * pi;  w11i[n] = 0.0f;               // conj(P)*P
  }
  __syncthreads();

  // ---- Cauchy kernel -> atRoots in S ----
  for (int l = tid; l < 4096; l += NTHREADS) {
    float phi = -2.0f * PI_F * (float)l / 4096.0f;
    float so, co; __sincosf(phi, &so, &co);          // Omega = (co, so)
    float dr = 1.0f + co, di = so;                   // 1+Omega
    float inv = 1.0f / (dr * dr + di * di);
    float i1r = dr * inv, i1i = -di * inv;           // 1/(1+Omega)
    float clr = 2.0f * i1r, cli = 2.0f * i1i;        // c
    float nr = 1.0f - co, ni = -so;                  // 1-Omega
    float sc2 = 2.0f / step;
    float gr = (nr * i1r - ni * i1i) * sc2;
    float gi = (nr * i1i + ni * i1r) * sc2;          // g
    float k00r = 0, k00i = 0, k01r = 0, k01i = 0;
    float k10r = 0, k10i = 0, k11r = 0, k11i = 0;
    for (int n = 0; n < 64; ++n) {
      float ddr = gr - lamr[n], ddi = gi - lami[n];
      float ir = 1.0f / (ddr * ddr + ddi * ddi);
      float vr = ddr * ir, vi = -ddi * ir;           // 1/(g-Lam)
      k00r += w00r[n] * vr - w00i[n] * vi;  k00i += w00r[n] * vi + w00i[n] * vr;
      k01r += w01r[n] * vr - w01i[n] * vi;  k01i += w01r[n] * vi + w01i[n] * vr;
      k10r += w10r[n] * vr - w10i[n] * vi;  k10i += w10r[n] * vi + w10i[n] * vr;
      k11r += w11r[n] * vr - w11i[n] * vi;  k11i += w11r[n] * vi + w11i[n] * vr;
    }
    float er = 1.0f + k11r, ei = k11i;
    float iv = 1.0f / (er * er + ei * ei);
    float i2r = er * iv, i2i = -ei * iv;             // 1/(1+k11)
    float m1r = k01r * i2r - k01i * i2i, m1i = k01r * i2i + k01i * i2r;
    float ttr = m1r * k10r - m1i * k10i, tti = m1r * k10i + m1i * k10r;
    float rr = k00r - ttr, ri = k00i - tti;
    Sr[l] = clr * rr - cli * ri;
    Si[l] = clr * ri + cli * rr;
  }

  // ---- K = ifft(atRoots): S -> A (S preserved) ----
  fft4096(Sr, Si, Br, Bi, Ar, Ai, Fr, Fi, -1.0f, 1.0f / 4096.0f, tid);

  // ---- Kmod = Re(K) * w,  w[l] = exp(-i pi l/4096) ----
  for (int l = tid; l < 4096; l += NTHREADS) {
    float ws, wc; __sincosf(-PI_F * (float)l / 4096.0f, &ws, &wc);
    float kr = Ar[l];
    Ar[l] = kr * wc; Ai[l] = kr * ws;
  }
  // ---- Kodd = fft(Kmod): A -> A  (odd 2L-bins of K spectrum) ----
  fft4096(Ar, Ai, Br, Bi, Ar, Ai, Fr, Fi, +1.0f, 1.0f, tid);

  // ---- Ke (even bins) = Hermitian part of atRoots, in place ----
  for (int m = tid; m <= 2048; m += NTHREADS) {
    int p = (4096 - m) & 4095;
    float a0r = Sr[m], a0i = Si[m], b0r = Sr[p], b0i = Si[p];
    float kr = 0.5f * (a0r + b0r), ki = 0.5f * (a0i - b0i);
    Sr[m] = kr; Si[m] = ki;
    if (p != m) { Sr[p] = kr; Si[p] = -ki; }
  }

  // u must be resident in LDS from here on (TDM completion + barrier)
#if HAVE_TDM
  if (tid < 32) __builtin_amdgcn_s_wait_tensorcnt(0);
#endif
  __syncthreads();

  // ---- Uo = fft(u * w): T -> T ----
  for (int l = tid; l < 4096; l += NTHREADS) {
    float ws, wc; __sincosf(-PI_F * (float)l / 4096.0f, &ws, &wc);
    float uv = su[l];
    Tr[l] = uv * wc; Ti[l] = uv * ws;
  }
  fft4096(Tr, Ti, Br, Bi, Tr, Ti, Fr, Fi, +1.0f, 1.0f, tid);

  // ---- Po = Uo*Kodd -> T; yO = ifft(Po) stays in T ----
  for (int l = tid; l < 4096; l += NTHREADS) {
    float xr = Tr[l], xi = Ti[l], yr = Ar[l], yi = Ai[l];
    Tr[l] = xr * yr - xi * yi;
    Ti[l] = xr * yi + xi * yr;
  }
  fft4096(Tr, Ti, Br, Bi, Tr, Ti, Fr, Fi, -1.0f, 1.0f / 4096.0f, tid);

  // ---- Ue = fft(u): A -> A ----
  for (int l = tid; l < 4096; l += NTHREADS) { Ar[l] = su[l]; Ai[l] = 0.0f; }
  fft4096(Ar, Ai, Br, Bi, Ar, Ai, Fr, Fi, +1.0f, 1.0f, tid);

  // ---- Pe = Ue*Ke -> A; yE = ifft(Pe) ----
  for (int l = tid; l < 4096; l += NTHREADS) {
    float xr = Ar[l], xi = Ai[l], yr = Sr[l], yi = Si[l];
    Ar[l] = xr * yr - xi * yi;
    Ai[l] = xr * yi + xi * yr;
  }
  fft4096(Ar, Ai, Br, Bi, Ar, Ai, Fr, Fi, -1.0f, 1.0f / 4096.0f, tid);

  // ---- single fused store: y = 0.5*Re(yE) + 0.5*Re(conj(w)*yO) + D*u ----
  for (int l = tid; l < 4096; l += NTHREADS) {
    float ws, wc; __sincosf(-PI_F * (float)l / 4096.0f, &ws, &wc);
    oh[l] = 0.5f * Ar[l] + 0.5f * (wc * Tr[l] + ws * Ti[l]) + dcoef * su[l];
  }
}

extern "C" void kernel_launch(void* const* d_in, const int* in_sizes, int n_in,
                              void* d_out, int out_size, void* d_ws, size_t ws_size,
                              hipStream_t stream) {
  const float* u   = (const float*)d_in[0];
  const float* lre = (const float*)d_in[1];
  const float* lim = (const float*)d_in[2];
  const float* pre = (const float*)d_in[3];
  const float* pim = (const float*)d_in[4];
  const float* bre = (const float*)d_in[5];
  const float* bim = (const float*)d_in[6];
  const float* cm  = (const float*)d_in[7];
  const float* dm  = (const float*)d_in[8];
  const float* ls  = (const float*)d_in[9];
  float* out = (float*)d_out;
  const size_t shmem = (size_t)(11 * 4096 + 10 * 64) * sizeof(float); // 182,784 B
  s4_channel_kernel<<<dim3(512), dim3(NTHREADS), shmem, stream>>>(
      u, lre, lim, pre, pim, bre, bim, cm, dm, ls, out);
}